// SSIMMetric_16827681865845
// MI455X (gfx1250) — compile-verified
//
#include <hip/hip_runtime.h>
#include <stdint.h>

#define IMG_B 16
#define IMG_C 3
#define IMG_H 512
#define IMG_W 512
#define OUT_H 506
#define OUT_W 506
#define INV_NP (1.0f / 49.0f)
#define COV_NORM (49.0f / 48.0f)

// ---------------------------------------------------------------------------
// Kernel 1: per-image min/max partials of pred. 16 images x 64 blocks.
// Each block: 256 threads x 12 float4 = 3072 float4 = 12288 floats.
// ---------------------------------------------------------------------------
__global__ __launch_bounds__(256) void k_minmax(const float* __restrict__ pred,
                                                float* __restrict__ mnp,
                                                float* __restrict__ mxp) {
  int img = blockIdx.x >> 6;
  int blk = blockIdx.x & 63;
  const float4* p = (const float4*)(pred + (size_t)img * (IMG_C * IMG_H * IMG_W));
  int base = blk * 3072 + threadIdx.x;
  float mn = 1e30f, mx = -1e30f;
#pragma unroll
  for (int i = 0; i < 12; ++i) {
    float4 v = p[base + i * 256];
    mn = fminf(mn, fminf(fminf(v.x, v.y), fminf(v.z, v.w)));
    mx = fmaxf(mx, fmaxf(fmaxf(v.x, v.y), fmaxf(v.z, v.w)));
  }
  for (int off = 16; off > 0; off >>= 1) {
    mn = fminf(mn, __shfl_down(mn, off, 32));
    mx = fmaxf(mx, __shfl_down(mx, off, 32));
  }
  __shared__ float smn[8], smx[8];
  int lane = threadIdx.x & 31, wv = threadIdx.x >> 5;
  if (lane == 0) { smn[wv] = mn; smx[wv] = mx; }
  __syncthreads();
  if (threadIdx.x == 0) {
    mn = smn[0]; mx = smx[0];
    for (int i = 1; i < 8; ++i) { mn = fminf(mn, smn[i]); mx = fmaxf(mx, smx[i]); }
    mnp[blockIdx.x] = mn;
    mxp[blockIdx.x] = mx;
  }
}

// ---------------------------------------------------------------------------
// Kernel 2: reduce 64 partials per image -> c1, c2.
// ---------------------------------------------------------------------------
__global__ __launch_bounds__(64) void k_range(const float* __restrict__ mnp,
                                              const float* __restrict__ mxp,
                                              float* __restrict__ c12) {
  int img = blockIdx.x;
  float mn = mnp[img * 64 + threadIdx.x];
  float mx = mxp[img * 64 + threadIdx.x];
  for (int off = 16; off > 0; off >>= 1) {
    mn = fminf(mn, __shfl_down(mn, off, 32));
    mx = fmaxf(mx, __shfl_down(mx, off, 32));
  }
  __shared__ float smn[2], smx[2];
  int lane = threadIdx.x & 31, wv = threadIdx.x >> 5;
  if (lane == 0) { smn[wv] = mn; smx[wv] = mx; }
  __syncthreads();
  if (threadIdx.x == 0) {
    float dr = fmaxf(smx[0], smx[1]) - fminf(smn[0], smn[1]);
    float c1 = 0.01f * dr; c1 *= c1;
    float c2 = 0.03f * dr; c2 *= c2;
    c12[img * 2 + 0] = c1;
    c12[img * 2 + 1] = c2;
  }
}

// ---------------------------------------------------------------------------
// Kernel 3: SSIM tile kernel. One block = one 32x32 output tile of one
// (image, channel). Grid = 16*3*16*16 = 12288 blocks, contiguous per image.
// CDNA5 path: 16B-per-lane async DMA (global_load_async_to_lds_b128) of the
// 38x40(padded) input tiles, tracked on ASYNCcnt.
// ---------------------------------------------------------------------------
__global__ __launch_bounds__(256) void k_ssim(const float* __restrict__ tr,
                                              const float* __restrict__ pr,
                                              const float* __restrict__ c12,
                                              float* __restrict__ partial) {
  // T, P, TT, PP, TP tiles: 38 rows x 40 cols (10 x b128 chunks per row).
  __shared__ alignas(16) float tiles[5][38 * 40];
  __shared__ float vs[5][32 * 40];     // vertical 7-sums at (oy, x)
  __shared__ float hs[5][32 * 32];     // full 7x7 window sums at (oy, ox)
  __shared__ float red[8];

  const int bid = blockIdx.x;
  const int t   = bid & 255;       // tile index within channel
  const int bc  = bid >> 8;        // (img, channel)
  const int ch  = bc % IMG_C;
  const int img = bc / IMG_C;
  const int ty = t >> 4, tx = t & 15;
  const int y0 = ty * 32, x0 = tx * 32;
  const size_t chanOff = ((size_t)img * IMG_C + ch) * (IMG_H * IMG_W);
  const float* tBase = tr + chanOff;
  const float* pBase = pr + chanOff;
  const int tid = threadIdx.x;

  // ---- Phase A: tile DMA. 38 rows x 10 b128-chunks per tensor.
  // Edge tiles: chunk-clamped loads land defined (duplicate) data in the
  // out-of-image region; those cells only feed outputs masked in Phase D.
  for (int i = tid; i < 380; i += 256) {
    int r   = i / 10;
    int ch4 = (i % 10) * 4;                       // column of 4-float chunk
    int gy = y0 + r;   if (gy > IMG_H - 1) gy = IMG_H - 1;
    int gx = x0 + ch4; if (gx > IMG_W - 4) gx = IMG_W - 4;   // keep 16B chunk in-range
    size_t off = (size_t)gy * IMG_W + gx;
#if defined(__gfx1250__)
    uint32_t ldsT = (uint32_t)(size_t)&tiles[0][r * 40 + ch4];
    uint32_t ldsP = ldsT + (uint32_t)(38 * 40 * sizeof(float));
    // CDNA5 async DMA: copies 16 bytes per lane from the global address in vaddr to the LDS address in vdst; ASYNCcnt.
    asm volatile("global_load_async_to_lds_b128 %0, %1, off"
                 :: "v"(ldsT), "v"(tBase + off) : "memory");
    asm volatile("global_load_async_to_lds_b128 %0, %1, off"
                 :: "v"(ldsP), "v"(pBase + off) : "memory");
#else
    for (int k = 0; k < 4; ++k) {
      tiles[0][r * 40 + ch4 + k] = tBase[off + k];
      tiles[1][r * 40 + ch4 + k] = pBase[off + k];
    }
#endif
  }
#if defined(__gfx1250__)
#if __has_builtin(__builtin_amdgcn_s_wait_asynccnt)
  __builtin_amdgcn_s_wait_asynccnt(0);
#else
  asm volatile("s_wait_asynccnt 0" ::: "memory");
#endif
#endif
  __syncthreads();

  // ---- Phase A2: elementwise products into LDS
  for (int i = tid; i < 38 * 38; i += 256) {
    int r = i / 38, col = i % 38;
    float a = tiles[0][r * 40 + col];
    float b = tiles[1][r * 40 + col];
    tiles[2][r * 40 + col] = a * a;
    tiles[3][r * 40 + col] = b * b;
    tiles[4][r * 40 + col] = a * b;
  }
  __syncthreads();

  // ---- Phase B: vertical sliding 7-sums. Task = (x in 0..37, q in 0..4).
  if (tid < 190) {
    int x = tid % 38, q = tid / 38;
    const float* src = tiles[q];
    float acc = 0.0f;
#pragma unroll
    for (int k = 0; k < 7; ++k) acc += src[k * 40 + x];
    vs[q][x] = acc;
    for (int oy = 1; oy < 32; ++oy) {
      acc += src[(oy + 6) * 40 + x] - src[(oy - 1) * 40 + x];
      vs[q][oy * 40 + x] = acc;
    }
  }
  __syncthreads();

  // ---- Phase C: horizontal sliding 7-sums. Task = (oy in 0..31, q in 0..4).
  if (tid < 160) {
    int oy = tid % 32, q = tid / 32;
    const float* src = &vs[q][oy * 40];
    float acc = 0.0f;
#pragma unroll
    for (int k = 0; k < 7; ++k) acc += src[k];
    hs[q][oy * 32] = acc;
    for (int ox = 1; ox < 32; ++ox) {
      acc += src[ox + 6] - src[ox - 1];
      hs[q][oy * 32 + ox] = acc;
    }
  }
  __syncthreads();

  // ---- Phase D: SSIM formula + fixed-order block reduction
  const float c1 = c12[img * 2 + 0];
  const float c2 = c12[img * 2 + 1];
  int oh = OUT_H - y0; if (oh > 32) oh = 32;
  int ow = OUT_W - x0; if (ow > 32) ow = 32;
  float lsum = 0.0f;
  for (int i = tid; i < 1024; i += 256) {
    int oy = i >> 5, ox = i & 31;
    if (oy < oh && ox < ow) {
      float ux  = hs[0][i] * INV_NP;
      float uy  = hs[1][i] * INV_NP;
      float vx  = COV_NORM * (hs[2][i] * INV_NP - ux * ux);
      float vy  = COV_NORM * (hs[3][i] * INV_NP - uy * uy);
      float vxy = COV_NORM * (hs[4][i] * INV_NP - ux * uy);
      float a1 = 2.0f * ux * uy + c1;
      float a2 = 2.0f * vxy + c2;
      float b1 = ux * ux + uy * uy + c1;
      float b2 = vx + vy + c2;
      lsum += (a1 * a2) / (b1 * b2);
    }
  }
  for (int off = 16; off > 0; off >>= 1) lsum += __shfl_down(lsum, off, 32);
  int lane = tid & 31, wv = tid >> 5;
  if (lane == 0) red[wv] = lsum;
  __syncthreads();
  if (tid == 0) {
    float s = red[0];
    for (int i = 1; i < 8; ++i) s += red[i];
    partial[bid] = s;
  }
}

// ---------------------------------------------------------------------------
// Kernel 4: per-image final sum of 768 tile partials (fixed order).
// ---------------------------------------------------------------------------
__global__ __launch_bounds__(256) void k_final(const float* __restrict__ partial,
                                               float* __restrict__ out) {
  int img = blockIdx.x;
  const float* p = partial + img * 768;
  float s = 0.0f;
  for (int i = threadIdx.x; i < 768; i += 256) s += p[i];
  for (int off = 16; off > 0; off >>= 1) s += __shfl_down(s, off, 32);
  __shared__ float red[8];
  int lane = threadIdx.x & 31, wv = threadIdx.x >> 5;
  if (lane == 0) red[wv] = s;
  __syncthreads();
  if (threadIdx.x == 0) {
    float tot = red[0];
    for (int i = 1; i < 8; ++i) tot += red[i];
    out[img] = tot / (float)(IMG_C * OUT_H * OUT_W);
  }
}

// ---------------------------------------------------------------------------
// Launcher. Workspace layout (floats):
//   [0,1024)      per-block min partials
//   [1024,2048)   per-block max partials
//   [2048,2080)   (c1,c2) per image
//   [2080,14368)  per-tile SSIM partial sums (12288)
// ---------------------------------------------------------------------------
extern "C" void kernel_launch(void* const* d_in, const int* in_sizes, int n_in,
                              void* d_out, int out_size, void* d_ws, size_t ws_size,
                              hipStream_t stream) {
  const float* tr = (const float*)d_in[0];
  const float* pr = (const float*)d_in[1];
  float* ws      = (float*)d_ws;
  float* mnp     = ws;
  float* mxp     = ws + 1024;
  float* c12     = ws + 2048;
  float* partial = ws + 2080;
  float* out     = (float*)d_out;

  hipLaunchKernelGGL(k_minmax, dim3(IMG_B * 64), dim3(256), 0, stream, pr, mnp, mxp);
  hipLaunchKernelGGL(k_range,  dim3(IMG_B),      dim3(64),  0, stream, mnp, mxp, c12);
  hipLaunchKernelGGL(k_ssim,   dim3(IMG_B * IMG_C * 256), dim3(256), 0, stream,
                     tr, pr, c12, partial);
  hipLaunchKernelGGL(k_final,  dim3(IMG_B),      dim3(256), 0, stream, partial, out);
}